// ListMLELoss_36532991819989
// MI455X (gfx1250) — compile-verified
//
#include <hip/hip_runtime.h>

// ListMLE loss for [B, 2048] rows on gfx1250 (MI455X).
// One workgroup (256 threads = 8 wave32) per row:
//   1. async-stage predictions row into LDS (global_load_async_to_lds_b128,
//      ASYNCcnt-tracked) -- overlaps with everything up to the gather
//   2. build 64-bit stable sort keys from targets (two b128 loads/thread),
//      hybrid register/LDS bitonic sort: strides j<=4 in registers,
//      only j>=8 passes touch LDS (66 -> 36 LDS passes)
//   3. keys for the final chunk end in registers; gather preds by sorted
//      index from LDS, exp, chunked suffix-scan for the P-L denominator
//   4. per-row loss via fixed-tree reduction -> d_ws
// Second kernel: deterministic tree-reduce of row losses -> mean -> d_out.

#define L        2048
#define NTHREADS 256
#define CHUNK    (L / NTHREADS)   // 8 consecutive elements per thread

__global__ __launch_bounds__(NTHREADS) void listmle_row_kernel(
    const float* __restrict__ preds,
    const float* __restrict__ targets,
    float* __restrict__ row_loss)
{
    __shared__ unsigned long long skeys[L];          // 16 KB
    __shared__ __align__(16) float spreds[L];        //  8 KB
    __shared__ float partials[NTHREADS];             //  1 KB
    __shared__ float red[NTHREADS];                  //  1 KB

    const int tid  = threadIdx.x;
    const int row  = blockIdx.x;
    const int base = tid * CHUNK;
    const float* __restrict__ gp = preds   + (size_t)row * L;
    const float* __restrict__ gt = targets + (size_t)row * L;

    // ---- 1. Async global->LDS stage of the predictions row (B128 = 4 floats/lane).
    // LDS dest address for the async unit is the workgroup-relative byte offset,
    // i.e. the low 32 bits of the generic pointer to __shared__ data.
    {
        unsigned ldsBase = (unsigned)(unsigned long long)(&spreds[0]);
        #pragma unroll
        for (int it = 0; it < L / (NTHREADS * 4); ++it) {     // 2 iterations
            unsigned elem   = (unsigned)(it * NTHREADS + tid) * 4u;
            unsigned ldsOff = ldsBase + elem * 4u;
            unsigned long long gaddr = (unsigned long long)(gp + elem);
            asm volatile("global_load_async_to_lds_b128 %0, %1, off"
                         :: "v"(ldsOff), "v"(gaddr)
                         : "memory");
        }
    }

    // ---- 2. Build stable-descending sort keys for this thread's 8 elements.
    // orderedFloat: monotone float->uint map; ~ordered => descending target;
    // low 32 bits = original index => ties broken ascending (stable argsort(-t)).
    // Keys are globally distinct, so the non-stable bitonic network still
    // produces the exact stable order, deterministically.
    unsigned long long key[CHUNK];
    {
        const float4 t0 = ((const float4*)gt)[tid * 2 + 0];   // b128
        const float4 t1 = ((const float4*)gt)[tid * 2 + 1];   // b128
        const float tv[CHUNK] = { t0.x, t0.y, t0.z, t0.w, t1.x, t1.y, t1.z, t1.w };
        #pragma unroll
        for (int q = 0; q < CHUNK; ++q) {
            unsigned u   = __float_as_uint(tv[q]);
            unsigned ord = u ^ ((u >> 31) ? 0xFFFFFFFFu : 0x80000000u);
            key[q] = ((unsigned long long)(~ord) << 32) | (unsigned)(base + q);
        }
    }

    // ---- Bitonic stages k = 2, 4, 8: strides j <= 4 => entirely in registers.
    // base is 8-aligned, so (base+q)&k == q&k for k<8; for k==8 it is per-thread.
    #pragma unroll
    for (unsigned k = 2; k <= 8; k <<= 1) {
        #pragma unroll
        for (unsigned j = k >> 1; j > 0; j >>= 1) {
            #pragma unroll
            for (unsigned q = 0; q < CHUNK; ++q) {
                const unsigned qxj = q ^ j;
                if (qxj > q) {
                    const bool up = (((unsigned)(base + q) & k) == 0);
                    unsigned long long a = key[q], b = key[qxj];
                    if ((a > b) == up) { key[q] = b; key[qxj] = a; }
                }
            }
        }
    }
    #pragma unroll
    for (int q = 0; q < CHUNK; ++q) skeys[base + q] = key[q];
    __syncthreads();

    // ---- Bitonic stages k = 16 .. 2048: only strides j >= 8 go through LDS;
    // the j = 4,2,1 tail of each stage runs in registers with a uniform
    // direction bit ((base+q)&k is block-constant for k >= 16).
    for (unsigned k = 16; k <= (unsigned)L; k <<= 1) {
        for (unsigned j = k >> 1; j >= 8; j >>= 1) {
            for (unsigned i = tid; i < (unsigned)L; i += NTHREADS) {
                const unsigned ixj = i ^ j;
                if (ixj > i) {
                    unsigned long long a = skeys[i], b = skeys[ixj];
                    const bool up = ((i & k) == 0);
                    if ((a > b) == up) { skeys[i] = b; skeys[ixj] = a; }
                }
            }
            __syncthreads();
        }
        {
            const bool up = (((unsigned)base & k) == 0);
            #pragma unroll
            for (int q = 0; q < CHUNK; ++q) key[q] = skeys[base + q];
            #pragma unroll
            for (unsigned j = 4; j > 0; j >>= 1) {
                #pragma unroll
                for (unsigned q = 0; q < CHUNK; ++q) {
                    const unsigned qxj = q ^ j;
                    if (qxj > q) {
                        unsigned long long a = key[q], b = key[qxj];
                        if ((a > b) == up) { key[q] = b; key[qxj] = a; }
                    }
                }
            }
            if (k < (unsigned)L) {          // final stage keeps keys in registers
                #pragma unroll
                for (int q = 0; q < CHUNK; ++q) skeys[base + q] = key[q];
                __syncthreads();
            }
        }
    }

    // ---- Async preds must have landed before the gather (wave-local wait,
    // then barrier so every wave's async writes are visible to all readers).
    asm volatile("s_wait_asynccnt 0" ::: "memory");
    __syncthreads();

    // ---- 3. Gather + exp + suffix sums. This thread's registers hold the
    // sorted keys for ranks [base, base+8): suffix within chunk sequentially,
    // cross-chunk via log-step suffix scan of chunk totals.
    float ev[CHUNK];
    float denomLoc[CHUNK];
    #pragma unroll
    for (int q = 0; q < CHUNK; ++q) {
        const unsigned idx = (unsigned)(key[q] & 0xFFFFFFFFull);
        ev[q] = expf(spreds[idx]);
    }
    float suf = 0.0f;
    #pragma unroll
    for (int q = CHUNK - 1; q >= 0; --q) { suf += ev[q]; denomLoc[q] = suf; }
    partials[tid] = suf;
    __syncthreads();

    for (int off = 1; off < NTHREADS; off <<= 1) {
        const float add = (tid + off < NTHREADS) ? partials[tid + off] : 0.0f;
        __syncthreads();
        partials[tid] += add;
        __syncthreads();
    }
    const float excl = (tid + 1 < NTHREADS) ? partials[tid + 1] : 0.0f;

    // ---- 4. Per-row loss (exact reference form, both epsilons included).
    float lsum = 0.0f;
    #pragma unroll
    for (int q = 0; q < CHUNK; ++q) {
        const float denom = denomLoc[q] + excl;
        const float P = ev[q] / (denom + 1e-10f);
        lsum -= logf(P + 1e-10f);
    }
    __syncthreads();
    red[tid] = lsum;
    __syncthreads();
    #pragma unroll
    for (int s = NTHREADS / 2; s > 0; s >>= 1) {
        if (tid < s) red[tid] += red[tid + s];
        __syncthreads();
    }
    if (tid == 0) row_loss[row] = red[0];
}

__global__ __launch_bounds__(256) void listmle_reduce_kernel(
    const float* __restrict__ row_loss, float* __restrict__ out, int B)
{
    __shared__ float red[256];
    const int tid = threadIdx.x;
    float s = 0.0f;
    for (int i = tid; i < B; i += 256) s += row_loss[i];   // fixed order
    red[tid] = s;
    __syncthreads();
    #pragma unroll
    for (int k = 128; k > 0; k >>= 1) {
        if (tid < k) red[tid] += red[tid + k];
        __syncthreads();
    }
    if (tid == 0) out[0] = red[0] * (1.0f / (float)B);
}

extern "C" void kernel_launch(void* const* d_in, const int* in_sizes, int n_in,
                              void* d_out, int out_size, void* d_ws, size_t ws_size,
                              hipStream_t stream) {
    const float* preds   = (const float*)d_in[0];
    const float* targets = (const float*)d_in[1];
    const int B = in_sizes[0] / L;          // 4096 for the reference shapes

    float* row_loss = (float*)d_ws;         // B floats of scratch

    listmle_row_kernel<<<B, NTHREADS, 0, stream>>>(preds, targets, row_loss);
    listmle_reduce_kernel<<<1, 256, 0, stream>>>(row_loss, (float*)d_out, B);
}